// DistAttn_37967510897369
// MI455X (gfx1250) — compile-verified
//
#include <hip/hip_runtime.h>
#include <hip/hip_bf16.h>
#include <math.h>

#define NN 40000
#define EE 640000
#define F 128
#define NTILES_ROW (NN / 16)   // 2500, exact
#define NTILES_COL (F / 16)    // 8

typedef __attribute__((ext_vector_type(2))) float v2f;
typedef __attribute__((ext_vector_type(8))) float v8f;

// ---------------------------------------------------------------------------
// Kernel 0: init — zero out, zero denom, emax = -inf (as raw bits)
// ---------------------------------------------------------------------------
__global__ void init_kernel(float* __restrict__ out, float* __restrict__ denom,
                            int* __restrict__ emax_bits) {
    int i = blockIdx.x * blockDim.x + threadIdx.x;
    if (i < NN * F) out[i] = 0.0f;
    if (i < NN) {
        denom[i] = 0.0f;
        emax_bits[i] = (int)0xFF800000u;   // -inf
    }
}

// ---------------------------------------------------------------------------
// Kernel 1: fused triple GEMM  H = feat@Wfc, Q = feat@Wq, K = feat@Wk
// One wave per 16x16 output tile, v_wmma_f32_16x16x4_f32, A fragment reused
// across the three B matrices.
// ---------------------------------------------------------------------------
__global__ void gemm3_wmma(const float* __restrict__ feat,
                           const float* __restrict__ Wfc,
                           const float* __restrict__ Wq,
                           const float* __restrict__ Wk,
                           float* __restrict__ H,
                           float* __restrict__ Q,
                           float* __restrict__ K) {
    const int wid  = (blockIdx.x * blockDim.x + threadIdx.x) >> 5;
    const int lane = threadIdx.x & 31;
    const int rowTile = wid >> 3;        // wid / NTILES_COL
    const int colTile = wid & 7;         // wid % NTILES_COL
    if (rowTile >= NTILES_ROW) return;   // uniform per wave: EXEC stays all-1

    const int halfSel = lane >> 4;       // 0: lanes 0-15, 1: lanes 16-31
    const int sub     = lane & 15;
    const int kOff    = halfSel * 2;     // A/B K-split across wave halves
    const int rowA    = rowTile * 16 + sub;          // A: M = lane (mod 16)
    const int colB    = colTile * 16 + sub;          // B: N = lane (mod 16)

    v8f accH = {};
    v8f accQ = {};
    v8f accK = {};

    const float* frow = feat + rowA * F;

    #pragma unroll 4
    for (int k0 = 0; k0 < F; k0 += 4) {
        const int ka = k0 + kOff;
        // A fragment: VGPR0 = A[M][ka], VGPR1 = A[M][ka+1]  (contiguous b64)
        v2f a;
        a.x = frow[ka];
        a.y = frow[ka + 1];
        // B fragments: row ka / ka+1 of each weight, N = colB
        const int w0 = ka * F + colB;
        v2f bfc, bq, bk;
        bfc.x = Wfc[w0]; bfc.y = Wfc[w0 + F];
        bq.x  = Wq[w0];  bq.y  = Wq[w0 + F];
        bk.x  = Wk[w0];  bk.y  = Wk[w0 + F];

        accH = __builtin_amdgcn_wmma_f32_16x16x4_f32(false, a, false, bfc,
                                                     (short)0, accH, false, false);
        accQ = __builtin_amdgcn_wmma_f32_16x16x4_f32(false, a, false, bq,
                                                     (short)0, accQ, false, false);
        accK = __builtin_amdgcn_wmma_f32_16x16x4_f32(false, a, false, bk,
                                                     (short)0, accK, false, false);
    }

    // D layout: VGPR j -> D[M = j + 8*halfSel][N = sub]
    const int outColBase = colTile * 16 + sub;
    const int outRowBase = rowTile * 16 + halfSel * 8;
    #pragma unroll
    for (int j = 0; j < 8; ++j) {
        const int off = (outRowBase + j) * F + outColBase;
        H[off] = accH[j];
        Q[off] = accQ[j];
        K[off] = accK[j];
    }
}

// ---------------------------------------------------------------------------
// Kernel 2: per-edge score e = dot(Q[dst], K[src]) / sqrt(128)
// + segmented max into emax via int-bit atomic trick. One wave per edge.
// ---------------------------------------------------------------------------
__device__ __forceinline__ void atomicMaxF(float* addr, float v) {
    if (v >= 0.0f)
        atomicMax((int*)addr, __float_as_int(v));
    else
        atomicMin((unsigned int*)addr, (unsigned int)__float_as_int(v));
}

__global__ void edge_scores(const float* __restrict__ Q,
                            const float* __restrict__ K,
                            const int* __restrict__ src,
                            const int* __restrict__ dst,
                            float* __restrict__ e,
                            float* __restrict__ emax) {
    const int wid  = (blockIdx.x * blockDim.x + threadIdx.x) >> 5;
    const int lane = threadIdx.x & 31;
    if (wid >= EE) return;
    const int s = src[wid];
    const int d = dst[wid];
    const float4 qv = ((const float4*)(Q + (size_t)d * F))[lane];
    const float4 kv = ((const float4*)(K + (size_t)s * F))[lane];
    float p = qv.x * kv.x + qv.y * kv.y + qv.z * kv.z + qv.w * kv.w;
    #pragma unroll
    for (int off = 16; off > 0; off >>= 1)
        p += __shfl_down(p, off, 32);
    if (lane == 0) {
        const float sc = p * 0.0883883476483184f;   // 1/sqrt(128)
        e[wid] = sc;
        atomicMaxF(&emax[d], sc);
    }
}

// ---------------------------------------------------------------------------
// Kernel 3: ex = exp(e - emax[dst]); denom[dst] += ex  (one thread per edge)
// ---------------------------------------------------------------------------
__global__ void edge_exp(const float* __restrict__ e,
                         const int* __restrict__ dst,
                         const float* __restrict__ emax,
                         float* __restrict__ ex,
                         float* __restrict__ denom) {
    const int i = blockIdx.x * blockDim.x + threadIdx.x;
    if (i >= EE) return;
    const int d = dst[i];
    const float v = __expf(e[i] - emax[d]);
    ex[i] = v;
    atomicAdd(&denom[d], v);
}

// ---------------------------------------------------------------------------
// Kernel 4: out[dst] += H[src] * (ex / denom[dst]). One wave per edge,
// 4 coalesced f32 atomics per lane.
// ---------------------------------------------------------------------------
__global__ void edge_scatter(const float* __restrict__ H,
                             const float* __restrict__ ex,
                             const float* __restrict__ denom,
                             const int* __restrict__ src,
                             const int* __restrict__ dst,
                             float* __restrict__ out) {
    const int wid  = (blockIdx.x * blockDim.x + threadIdx.x) >> 5;
    const int lane = threadIdx.x & 31;
    if (wid >= EE) return;
    const int s = src[wid];
    const int d = dst[wid];
    const float a = ex[wid] / denom[d];
    const float4 hv = ((const float4*)(H + (size_t)s * F))[lane];
    float* o = out + (size_t)d * F + lane * 4;
    atomicAdd(o + 0, hv.x * a);
    atomicAdd(o + 1, hv.y * a);
    atomicAdd(o + 2, hv.z * a);
    atomicAdd(o + 3, hv.w * a);
}

// ---------------------------------------------------------------------------
// Launch
// ---------------------------------------------------------------------------
extern "C" void kernel_launch(void* const* d_in, const int* in_sizes, int n_in,
                              void* d_out, int out_size, void* d_ws, size_t ws_size,
                              hipStream_t stream) {
    const float* feat = (const float*)d_in[0];
    // d_in[1] = loc (dead), d_in[5..9] = e2-branch params (dead)
    const float* Wfc  = (const float*)d_in[2];
    const float* Wq   = (const float*)d_in[3];
    const float* Wk   = (const float*)d_in[4];
    const int*   src  = (const int*)d_in[10];
    const int*   dst  = (const int*)d_in[11];
    float* out = (float*)d_out;

    // workspace layout
    char* ws = (char*)d_ws;
    const size_t NF = (size_t)NN * F * sizeof(float);       // 20.48 MB each
    float* H     = (float*)(ws);
    float* Q     = (float*)(ws + NF);
    float* K     = (float*)(ws + 2 * NF);
    float* ex    = (float*)(ws + 3 * NF);                   // E floats
    float* emax  = (float*)(ws + 3 * NF + (size_t)EE * 4);
    float* denom = (float*)(ws + 3 * NF + (size_t)EE * 4 + (size_t)NN * 4);

    // 0: init out / denom / emax
    {
        const int total = NN * F;
        init_kernel<<<(total + 255) / 256, 256, 0, stream>>>(out, denom, (int*)emax);
    }
    // 1: H,Q,K via WMMA — 20000 waves, 8 waves/block
    {
        const int waves = NTILES_ROW * NTILES_COL;
        gemm3_wmma<<<waves / 8, 256, 0, stream>>>(feat, Wfc, Wq, Wk, H, Q, K);
    }
    // 2: edge scores + segment max — 1 wave/edge
    {
        const int waves = EE;
        edge_scores<<<(waves * 32 + 255) / 256, 256, 0, stream>>>(Q, K, src, dst, ex, emax);
    }
    // 3: exp + denom — 1 thread/edge (ex reused in place: reads e, writes ex)
    edge_exp<<<(EE + 255) / 256, 256, 0, stream>>>(ex, dst, emax, ex, denom);
    // 4: weighted scatter — 1 wave/edge
    {
        const int waves = EE;
        edge_scatter<<<(waves * 32 + 255) / 256, 256, 0, stream>>>(H, ex, denom, src, dst, out);
    }
}